// GATDecoder_61280593379512
// MI455X (gfx1250) — compile-verified
//
#include <hip/hip_runtime.h>
#include <hip/hip_bf16.h>
#include <math.h>

// ---------------------------------------------------------------------------
// GAT 2-layer forward for MI455X (gfx1250, wave32).
// GEMMs: V_WMMA_F32_16X16X4_F32, B panel staged in LDS (reused by 8 waves),
// each wave computes a 16 x NC strip (8 WMMAs per A-load for layer 1).
// Edge phases are memory-bound scatter/gather; node state fits in 192MB L2.
// ---------------------------------------------------------------------------

typedef float v2f __attribute__((ext_vector_type(2)));
typedef float v8f __attribute__((ext_vector_type(8)));

#define NEG_SLOPE 0.2f

// ---------------- generic fill ----------------
__global__ void fill_k(float* __restrict__ p, long long n, float v) {
    long long t = (long long)blockIdx.x * blockDim.x + threadIdx.x;
    if (t < n) p[t] = v;
}

// ---------------- fp32 WMMA GEMM: C[M,NC] = A[M,K] @ B[K,NC] ----------------
// B panel cached in LDS; one wave computes a 16-row x NC strip.
// All guards are wave-uniform and after the barrier (EXEC all-1s at WMMA).
template <int K, int NC>
__global__ void gemm_wmma_f32_k(const float* __restrict__ A,
                                const float* __restrict__ B,
                                float* __restrict__ C, int M) {
    constexpr int NT = NC / 16;                 // 16-col tiles per strip
    __shared__ float Blds[K * NC];

    // cooperative, coalesced stage of the full B panel into LDS
    for (int i = threadIdx.x; i < K * NC; i += blockDim.x)
        Blds[i] = B[i];
    __syncthreads();

    int wave = (int)((blockIdx.x * blockDim.x + threadIdx.x) >> 5);
    int lane = threadIdx.x & 31;
    int mtiles = M >> 4;
    if (wave >= mtiles) return;                 // whole-wave uniform exit
    int row0 = wave << 4;

    int half = lane >> 4;                       // 0: lanes 0-15, 1: lanes 16-31
    int mrow = lane & 15;                       // A row (M) held by this lane
    int ncol = lane & 15;                       // B/C column within 16-tile
    int koff = half << 1;                       // K-pair offset: 0 or 2

    const float* Ar = A + (size_t)(row0 + mrow) * K + koff;

    v8f zero = {};
    v8f acc[NT];
#pragma unroll
    for (int j = 0; j < NT; ++j) acc[j] = zero;

#pragma unroll 4
    for (int k = 0; k < K; k += 4) {
        v2f a;
        a.x = Ar[k];                            // A[m][k+koff]
        a.y = Ar[k + 1];                        // A[m][k+koff+1]
        const float* B0 = &Blds[(k + koff) * NC + ncol];
#pragma unroll
        for (int j = 0; j < NT; ++j) {
            v2f b;
            b.x = B0[j * 16];                   // B[k+koff  ][col]
            b.y = B0[NC + j * 16];              // B[k+koff+1][col]
            acc[j] = __builtin_amdgcn_wmma_f32_16x16x4_f32(
                false, a, false, b, (short)0, acc[j], false, false);
        }
    }

    float* Cr = C + (size_t)(row0 + half * 8) * NC + ncol;
#pragma unroll
    for (int j = 0; j < NT; ++j) {
#pragma unroll
        for (int v = 0; v < 8; ++v)
            Cr[(size_t)v * NC + j * 16] = acc[j][v];   // row v+8*half, col j*16+ncol
    }
}

// ---------------- per-(node,head) attention logits ----------------
template <int H, int C>
__global__ void logits_k(const float* __restrict__ hbuf,
                         const float* __restrict__ asrc,
                         const float* __restrict__ adst,
                         float* __restrict__ als,
                         float* __restrict__ ald, int N) {
    long long t = (long long)blockIdx.x * blockDim.x + threadIdx.x;
    if (t >= (long long)N * H) return;
    int n = (int)(t / H);
    int h = (int)(t - (long long)n * H);
    const float* hp = hbuf + (size_t)n * (H * C) + (size_t)h * C;
    const float* as = asrc + (size_t)h * C;
    const float* ad = adst + (size_t)h * C;
    float ss = 0.f, sd = 0.f;
#pragma unroll
    for (int c = 0; c < C; ++c) {
        float v = hp[c];
        ss += v * as[c];
        sd += v * ad[c];
    }
    als[t] = ss;
    ald[t] = sd;
}

// float atomic max via monotone int/uint mapping (handles mixed signs)
__device__ __forceinline__ void atomicMaxF(float* addr, float v) {
    if (v >= 0.f)
        atomicMax((int*)addr, __float_as_int(v));
    else
        atomicMin((unsigned int*)addr, __float_as_uint(v));
}

// ---------------- edge pass A: leaky-relu logit + segment max ----------------
template <int H>
__global__ void edge_max_k(const int* __restrict__ src,
                           const int* __restrict__ dst,
                           const float* __restrict__ als,
                           const float* __restrict__ ald,
                           float* __restrict__ m,
                           float* __restrict__ ebuf, int E, int EP) {
    long long t = (long long)blockIdx.x * blockDim.x + threadIdx.x;
    if (t >= (long long)EP * H) return;
    int e = (int)(t / H);
    int h = (int)(t - (long long)e * H);
    int s, d;
    if (e < E) { s = src[e]; d = dst[e]; } else { s = d = e - E; }
    float x = als[(size_t)s * H + h] + ald[(size_t)d * H + h];
    x = (x > 0.f) ? x : NEG_SLOPE * x;
    ebuf[t] = x;
    atomicMaxF(&m[(size_t)d * H + h], x);
}

// ---------------- edge pass B: exp + segment sum ----------------
template <int H>
__global__ void edge_exp_k(const int* __restrict__ dst,
                           const float* __restrict__ m,
                           float* __restrict__ ebuf,
                           float* __restrict__ denom, int E, int EP) {
    long long t = (long long)blockIdx.x * blockDim.x + threadIdx.x;
    if (t >= (long long)EP * H) return;
    int e = (int)(t / H);
    int h = (int)(t - (long long)e * H);
    int d = (e < E) ? dst[e] : (e - E);
    float ex = expf(ebuf[t] - m[(size_t)d * H + h]);
    ebuf[t] = ex;
    atomicAdd(&denom[(size_t)d * H + h], ex);
}

// ---------------- edge pass C: alpha-weighted message scatter-add ----------------
template <int H, int C>
__global__ void edge_msg_k(const int* __restrict__ src,
                           const int* __restrict__ dst,
                           const float* __restrict__ ebuf,
                           const float* __restrict__ denom,
                           const float* __restrict__ hin,
                           float* __restrict__ out, int E, int EP) {
    const int HC = H * C;
    long long t = (long long)blockIdx.x * blockDim.x + threadIdx.x;
    if (t >= (long long)EP * HC) return;
    int e = (int)(t / HC);
    int r = (int)(t - (long long)e * HC);   // r = h*C + c
    int h = r / C;
    int s, d;
    if (e < E) { s = src[e]; d = dst[e]; } else { s = d = e - E; }
    float alpha = ebuf[(size_t)e * H + h] / denom[(size_t)d * H + h];
    atomicAdd(&out[(size_t)d * HC + r], hin[(size_t)s * HC + r] * alpha);
}

// ---------------- bias + ELU (layer-1 epilogue) ----------------
__global__ void bias_elu_k(float* __restrict__ p, const float* __restrict__ b,
                           long long n, int CH) {
    long long t = (long long)blockIdx.x * blockDim.x + threadIdx.x;
    if (t >= n) return;
    float v = p[t] + b[(int)(t % CH)];
    p[t] = (v > 0.f) ? v : expm1f(v);
}

// ---------------- bias add (layer-2 epilogue) ----------------
__global__ void bias_add_k(float* __restrict__ p, const float* __restrict__ b,
                           long long n, int CH) {
    long long t = (long long)blockIdx.x * blockDim.x + threadIdx.x;
    if (t >= n) return;
    p[t] = p[t] + b[(int)(t % CH)];
}

static inline int nblk(long long n, int b) { return (int)((n + b - 1) / b); }

extern "C" void kernel_launch(void* const* d_in, const int* in_sizes, int n_in,
                              void* d_out, int out_size, void* d_ws, size_t ws_size,
                              hipStream_t stream) {
    const float* x   = (const float*)d_in[0];
    const int*   ei  = (const int*)d_in[1];     // [2,E] int32
    const float* W1  = (const float*)d_in[2];   // [64,128]
    const float* as1 = (const float*)d_in[3];   // [4,32]
    const float* ad1 = (const float*)d_in[4];   // [4,32]
    const float* b1  = (const float*)d_in[5];   // [128]
    const float* W2  = (const float*)d_in[6];   // [128,32]
    const float* as2 = (const float*)d_in[7];   // [1,32]
    const float* ad2 = (const float*)d_in[8];   // [1,32]
    const float* b2  = (const float*)d_in[9];   // [32]
    float* out = (float*)d_out;

    const int N  = in_sizes[0] / 64;
    const int E  = in_sizes[1] / 2;
    const int EP = E + N;                       // edges + self-loops
    const int* src = ei;
    const int* dst = ei + E;

    // workspace layout (floats); layer-2 buffers alias dead layer-1 buffers
    float* ws = (float*)d_ws;
    size_t o = 0;
    float* h1   = ws + o; o += (size_t)N * 128;   // layer-1 features (h2 aliases)
    float* out1 = ws + o; o += (size_t)N * 128;   // layer-1 aggregation / layer-2 input
    float* als  = ws + o; o += (size_t)N * 4;
    float* ald  = ws + o; o += (size_t)N * 4;
    float* m    = ws + o; o += (size_t)N * 4;
    float* den  = ws + o; o += (size_t)N * 4;
    float* ebuf = ws + o; o += (size_t)EP * 4;    // per-edge logits / exp cache
    float* h2 = h1;                                // aliases (layer-1 dead by then)

    const int B = 256;

    // ---------------- layer 1 (H=4, C=32) ----------------
    fill_k<<<nblk((long long)N * 4, B), B, 0, stream>>>(m,   (long long)N * 4, -INFINITY);
    fill_k<<<nblk((long long)N * 4, B), B, 0, stream>>>(den, (long long)N * 4, 0.f);
    fill_k<<<nblk((long long)N * 128, B), B, 0, stream>>>(out1, (long long)N * 128, 0.f);

    {   // h1 = x @ W1   (M=N, NC=128, K=64): one wave per 16-row strip
        long long waves = (long long)(N / 16);
        gemm_wmma_f32_k<64, 128><<<nblk(waves * 32, B), B, 0, stream>>>(x, W1, h1, N);
    }
    logits_k<4, 32><<<nblk((long long)N * 4, B), B, 0, stream>>>(h1, as1, ad1, als, ald, N);
    edge_max_k<4><<<nblk((long long)EP * 4, B), B, 0, stream>>>(src, dst, als, ald, m, ebuf, E, EP);
    edge_exp_k<4><<<nblk((long long)EP * 4, B), B, 0, stream>>>(dst, m, ebuf, den, E, EP);
    edge_msg_k<4, 32><<<nblk((long long)EP * 128, B), B, 0, stream>>>(src, dst, ebuf, den, h1, out1, E, EP);
    bias_elu_k<<<nblk((long long)N * 128, B), B, 0, stream>>>(out1, b1, (long long)N * 128, 128);

    // ---------------- layer 2 (H=1, C=32) ----------------
    fill_k<<<nblk((long long)N, B), B, 0, stream>>>(m,   (long long)N, -INFINITY);
    fill_k<<<nblk((long long)N, B), B, 0, stream>>>(den, (long long)N, 0.f);
    fill_k<<<nblk((long long)N * 32, B), B, 0, stream>>>(out, (long long)N * 32, 0.f);

    {   // h2 = elu(out1) @ W2   (M=N, NC=32, K=128)
        long long waves = (long long)(N / 16);
        gemm_wmma_f32_k<128, 32><<<nblk(waves * 32, B), B, 0, stream>>>(out1, W2, h2, N);
    }
    logits_k<1, 32><<<nblk((long long)N, B), B, 0, stream>>>(h2, as2, ad2, als, ald, N);
    edge_max_k<1><<<nblk((long long)EP, B), B, 0, stream>>>(src, dst, als, ald, m, ebuf, E, EP);
    edge_exp_k<1><<<nblk((long long)EP, B), B, 0, stream>>>(dst, m, ebuf, den, E, EP);
    edge_msg_k<1, 32><<<nblk((long long)EP * 32, B), B, 0, stream>>>(src, dst, ebuf, den, h2, out, E, EP);
    bias_add_k<<<nblk((long long)N * 32, B), B, 0, stream>>>(out, b2, (long long)N * 32, 32);
}